// FixedMaskAdder_39264591020322
// MI455X (gfx1250) — compile-verified
//
#include <hip/hip_runtime.h>

#define NBATCH 16
#define LSEQ   2048
#define TILE   16
#define NTT    (LSEQ / TILE)

typedef __attribute__((ext_vector_type(2))) float v2f;
typedef __attribute__((ext_vector_type(8))) float v8f;

// Softmax is computed in BASE 2: log2(e) is folded into the q scaling so the
// WMMA directly produces base-2 logits and every exp is a bare v_exp_f32.
#define LOG2E_D 1.44269504088896340736
// reference's -1e9 additive causal mask, in the base-2 logit domain
#define MASK2 (-1.4426950408889634e9f)

__device__ __forceinline__ float exp2_fast(float x) {
  return __builtin_amdgcn_exp2f(x);   // v_exp_f32 (native exp2)
}

// ---------------------------------------------------------------------------
// Kernel 1: per-position precompute.
// For every (b, pos): token -> h -> rms_norm -> q/k/v heads -> rms_norm ->
// RoPE, mirroring the reference op-for-op in f32.  ATTN_SCALE * log2(e) is
// folded into q.  Derived constants are computed in f64 like the Python
// header.  Outputs (all L2-resident, 640 KB total):
//   qarr[b*L+t] = ATTN_SCALE*log2(e) * RoPE(rmsnorm(q))   (float2)
//   karr[b*L+s] = RoPE(rmsnorm(k))                        (float2)
//   varr[b*L+s] = hn1 * v         (the only nonzero V col) (float)
// ---------------------------------------------------------------------------
__global__ __launch_bounds__(256) void rope_precompute_kernel(
    const int* __restrict__ tokens, const float* __restrict__ Cp,
    const float* __restrict__ epsp, const float* __restrict__ vp,
    float2* __restrict__ qarr, float2* __restrict__ karr,
    float* __restrict__ varr) {
  int idx = blockIdx.x * blockDim.x + threadIdx.x;
  if (idx >= NBATCH * LSEQ) return;
  int pos = idx & (LSEQ - 1);

  // ---- derived constants (f64, as in the reference module header) ----
  const double PI    = 3.14159265358979323846;
  const double OMEGA = 2.0 * PI / 19.0;
  const double PHI   = OMEGA * (10.0 + 0.3);
  const double AMP   = log(10.0) / (cos(OMEGA * 0.3) - cos(OMEGA * (1.0 - 0.3)));
  const double QKS   = sqrt(AMP / sqrt(2.0));
  const float scale2 = (float)(pow(2.0, -0.5) * QKS * QKS * LOG2E_D);
  const float cos_p  = (float)cos(PHI);
  const float sin_p  = (float)sin(PHI);

  const float c  = Cp[0];
  const float ee = epsp[0];
  const float vs = vp[0];

  const int   d  = tokens[idx];
  const float df = (float)d;
  const float h0 = c - (ee * df) * df;   // tab[d,0]
  const float h1 = -df;                  // tab[d,1]

  // first rms_norm
  const float r   = rsqrtf((h0 * h0 + h1 * h1) * 0.5f + 1e-6f);
  const float hn0 = h0 * r;
  const float hn1 = h1 * r;

  // q head: rotate by -phi, then rms_norm
  float q0 = hn0 * cos_p;
  float q1 = hn0 * (-sin_p);
  const float qr = rsqrtf((q0 * q0 + q1 * q1) * 0.5f + 1e-6f);
  q0 *= qr; q1 *= qr;

  // k head: (hn0, 0), then rms_norm
  float k0 = hn0;
  const float kr = rsqrtf((k0 * k0) * 0.5f + 1e-6f);
  k0 *= kr;

  // RoPE at this position (theta computed in f32 like arange(L,f32)*OMEGA)
  const float theta = (float)pos * (float)OMEGA;
  const float ct = cosf(theta);
  const float st = sinf(theta);
  const float Qx = q0 * ct - q1 * st;
  const float Qy = q0 * st + q1 * ct;
  const float Kx = k0 * ct;            // k1 == 0
  const float Ky = k0 * st;

  qarr[idx] = make_float2(Qx * scale2, Qy * scale2);
  karr[idx] = make_float2(Kx, Ky);
  varr[idx] = hn1 * vs;
}

// ---------------------------------------------------------------------------
// Kernel 2: fused causal flash-attention (WMMA f32 16x16x4) + epilogue.
// One wave32 per (batch, 16-wide t-tile).  Scores are computed TRANSPOSED:
//   S'[s, t] = k_s . (ATTN_SCALE*log2e * q_t)  via  V_WMMA_F32_16X16X4_F32
// C/D layout: VGPR g, lane l -> (s = g + 8*(l>>4), t = l&15): each lane owns
// one t-column and 8 of its 16 s-values.  Each 16-lane half keeps an
// INDEPENDENT base-2 online-softmax triple (m, l, acc) over its own s-half
// (no cross-lane traffic in the loop), merged once at the end with the
// two-way rescale.  The masked diagonal tile is peeled.  Lanes 0..15 then
// run residual + gating + final-norm + 10-logit projection for their t.
// ---------------------------------------------------------------------------
__global__ __launch_bounds__(32) void flash_attn_epilogue_kernel(
    const int* __restrict__ tokens, const float* __restrict__ Cp,
    const float* __restrict__ epsp, const float* __restrict__ o_scalep,
    const float* __restrict__ g_basep, const float* __restrict__ g_slopep,
    const float* __restrict__ carry_ampp,
    const float2* __restrict__ qarr, const float2* __restrict__ karr,
    const float* __restrict__ varr, float* __restrict__ out) {
  const int lane = threadIdx.x;          // 0..31 (wave32)
  const int tt   = blockIdx.x & (NTT - 1);
  const int b    = blockIdx.x / NTT;
  const int col  = lane & 15;            // this lane's t within the tile
  const int hi   = lane >> 4;            // which s-half this lane covers
  const int tbase   = tt * TILE;
  const int rowbase = b * LSEQ;
  const bool lo_half = (hi == 0);

  // B fragment: scaled q for the 16 t-columns (K=2,3 slots zero).
  // Unconditional coalesced load at col; hi-half zeroed via v_cndmask.
  const float2 qld = qarr[rowbase + tbase + col];
  v2f bfrag;
  bfrag.x = lo_half ? qld.x : 0.0f;
  bfrag.y = lo_half ? qld.y : 0.0f;

  const float2* kp = karr + rowbase + col;                      // + j*TILE
  const float4* vp = (const float4*)(varr + rowbase + hi * 8);  // + j*4

  // per-half online softmax state (base-2 logit domain)
  float m    = -3.0e38f;
  float lsum = 0.0f;
  float osum = 0.0f;

  // ---- hot loop: all fully-unmasked s-tiles (j < tt), branch-free ----
  for (int j = 0; j < tt; ++j) {
    const float2 kld = kp[j * TILE];
    v2f afrag;
    afrag.x = lo_half ? kld.x : 0.0f;
    afrag.y = lo_half ? kld.y : 0.0f;
    __builtin_prefetch((const void*)(kp + (j + 1) * TILE), 0, 3);

    const float4 va = vp[j * 4];
    const float4 vb = vp[j * 4 + 1];
    const float vvv[8] = {va.x, va.y, va.z, va.w, vb.x, vb.y, vb.z, vb.w};

    v8f s = {};
    s = __builtin_amdgcn_wmma_f32_16x16x4_f32(
        /*neg_a=*/false, afrag, /*neg_b=*/false, bfrag,
        /*c_mod=*/(short)0, s, /*reuse_a=*/false, /*reuse_b=*/false);

    float tmax = s[0];
#pragma unroll
    for (int g = 1; g < 8; ++g) tmax = fmaxf(tmax, s[g]);
    const float mnew  = fmaxf(m, tmax);
    const float scale = exp2_fast(m - mnew);

    float le = 0.0f, oe = 0.0f;
#pragma unroll
    for (int g = 0; g < 8; ++g) {
      const float e = exp2_fast(s[g] - mnew);
      le += e;
      oe = fmaf(e, vvv[g], oe);
    }
    lsum = fmaf(lsum, scale, le);
    osum = fmaf(osum, scale, oe);
    m = mnew;
  }

  // ---- peeled diagonal tile (j == tt) with the -1e9*log2e causal mask ----
  {
    const float2 kld = kp[tt * TILE];
    v2f afrag;
    afrag.x = lo_half ? kld.x : 0.0f;
    afrag.y = lo_half ? kld.y : 0.0f;

    const float4 va = vp[tt * 4];
    const float4 vb = vp[tt * 4 + 1];
    const float vvv[8] = {va.x, va.y, va.z, va.w, vb.x, vb.y, vb.z, vb.w};

    v8f s = {};
    s = __builtin_amdgcn_wmma_f32_16x16x4_f32(
        false, afrag, false, bfrag, (short)0, s, false, false);

#pragma unroll
    for (int g = 0; g < 8; ++g)
      if ((g + hi * 8) > col) s[g] += MASK2;

    float tmax = s[0];
#pragma unroll
    for (int g = 1; g < 8; ++g) tmax = fmaxf(tmax, s[g]);
    const float mnew  = fmaxf(m, tmax);
    const float scale = exp2_fast(m - mnew);

    float le = 0.0f, oe = 0.0f;
#pragma unroll
    for (int g = 0; g < 8; ++g) {
      const float e = exp2_fast(s[g] - mnew);
      le += e;
      oe = fmaf(e, vvv[g], oe);
    }
    lsum = fmaf(lsum, scale, le);
    osum = fmaf(osum, scale, oe);
    m = mnew;
  }

  // ---- merge the two independent s-halves (3 shuffles, once per wave) ----
  const float m2 = __shfl_xor(m, 16, 32);
  const float l2 = __shfl_xor(lsum, 16, 32);
  const float o2 = __shfl_xor(osum, 16, 32);
  const float M  = fmaxf(m, m2);
  const float w1 = exp2_fast(m - M);
  const float w2 = exp2_fast(m2 - M);
  const float lt = fmaf(lsum, w1, l2 * w2);
  const float ot = fmaf(osum, w1, o2 * w2);
  const float attn0 = ot / lt;   // softmax-normalized single live V column

  if (lane < 16) {
    const int   t  = tbase + lane;
    const float c  = Cp[0];
    const float ee = epsp[0];
    const int   d  = tokens[rowbase + t];
    const float df = (float)d;
    const float h0 = c - (ee * df) * df;
    float       h1 = -df;

    // attention residual (only second channel is touched)
    h1 += o_scalep[0] * attn0;

    // rms_norm -> gated carry
    const float r   = rsqrtf((h0 * h0 + h1 * h1) * 0.5f + 1e-6f);
    const float hn0 = h0 * r;
    const float hn1 = h1 * r;
    const float a   = g_basep[0];
    const float gc  = g_slopep[0];
    const float g0  = hn0 * a + hn1 * gc;
    const float g1  = hn0 * (a - gc / c) + hn1 * gc;
    const float carry =
        carry_ampp[0] * (fmaxf(g1, 0.0f) * hn0 - fmaxf(g0, 0.0f) * hn0);
    h1 += carry;

    // final norm * nw, then logits = f @ tab.T
    const float r2  = rsqrtf((h0 * h0 + h1 * h1) * 0.5f + 1e-6f);
    const float inv_cn = 0.70710678118654752440f;  // 1/sqrt(MODEL_DIM)
    const float f0 = h0 * r2 * (0.1f * c * inv_cn);
    const float f1 = h1 * r2 * (-c * (1.0f / 50.0f) * inv_cn);

    float* op = out + (size_t)(rowbase + t) * 10;
#pragma unroll
    for (int jj = 0; jj < 10; ++jj) {
      const float dj = (float)jj;
      op[jj] = f0 * (c - (ee * dj) * dj) + f1 * (-dj);
    }
  }
}

extern "C" void kernel_launch(void* const* d_in, const int* in_sizes, int n_in,
                              void* d_out, int out_size, void* d_ws, size_t ws_size,
                              hipStream_t stream) {
  const int*   tokens    = (const int*)d_in[0];
  const float* C         = (const float*)d_in[1];
  const float* eps       = (const float*)d_in[2];
  const float* v         = (const float*)d_in[3];
  const float* o_scale   = (const float*)d_in[4];
  const float* g_base    = (const float*)d_in[5];
  const float* g_slope   = (const float*)d_in[6];
  const float* carry_amp = (const float*)d_in[7];
  float* out = (float*)d_out;

  // workspace partition (640 KB, L2-resident): qarr | karr | varr
  char* ws = (char*)d_ws;
  float2* qarr = (float2*)(ws);
  float2* karr = (float2*)(ws + 256 * 1024);
  float*  varr = (float*)(ws + 512 * 1024);

  const int total = NBATCH * LSEQ;
  rope_precompute_kernel<<<(total + 255) / 256, 256, 0, stream>>>(
      tokens, C, eps, v, qarr, karr, varr);

  flash_attn_epilogue_kernel<<<NBATCH * NTT, 32, 0, stream>>>(
      tokens, C, eps, o_scale, g_base, g_slope, carry_amp,
      qarr, karr, varr, out);
}